// FastGuidedFilter_58506044506586
// MI455X (gfx1250) — compile-verified
//
#include <hip/hip_runtime.h>
#include <hip/hip_bf16.h>
#include <stdint.h>
#include <stddef.h>

// ---------------------------------------------------------------------------
// FastGuidedFilter on gfx1250 (MI455X).
//   K1: low-res 5x5 box stats -> A,b. Horizontal box pass done on the matrix
//       unit as banded matmul: out(16x16) = V(16x20) x S(20x16) via 5 chained
//       V_WMMA_F32_16X16X4_F32 per quantity (exact f32 arithmetic).
//   K2: per 64x64 hi-res tile: async-stage 18x18 A/b region to LDS
//       (GLOBAL_LOAD_ASYNC_TO_LDS_B32, s_wait_asynccnt), then fused bilinear
//       (align_corners) + apply with float4 non-temporal streaming.
// ---------------------------------------------------------------------------

typedef __attribute__((ext_vector_type(2))) float v2f;
typedef __attribute__((ext_vector_type(4))) float f4;
typedef __attribute__((ext_vector_type(8))) float v8f;

#define NPLANE 24        // 8 batch * 3 channels
#define HL 256
#define WL 256
#define HH 1024
#define WH 1024
#define RAD 2
#define EPSV 1e-8f

#if defined(__gfx1250__) && __has_builtin(__builtin_amdgcn_wmma_f32_16x16x4_f32)
#define HAVE_WMMA_F32 1
#else
#define HAVE_WMMA_F32 0
#endif

#if defined(__gfx1250__) && __has_builtin(__builtin_amdgcn_global_load_async_to_lds_b32)
#define HAVE_ASYNC_LDS 1
#else
#define HAVE_ASYNC_LDS 0
#endif

__device__ __forceinline__ int imin_(int a, int b) { return a < b ? a : b; }
__device__ __forceinline__ int imax_(int a, int b) { return a > b ? a : b; }

// ---------------------------------------------------------------------------
// Kernel 1: low-res guided-filter coefficients. One wave (32 threads) per
// 16x16 output tile. Grid (16,16,24).
// ---------------------------------------------------------------------------
__global__ __launch_bounds__(32) void gf_lowres(const float* __restrict__ xlr,
                                                const float* __restrict__ ylr,
                                                float* __restrict__ Aout,
                                                float* __restrict__ Bout) {
  __shared__ float xs[20][21];
  __shared__ float ys[20][21];
  __shared__ float vx[16][20];
  __shared__ float vy[16][20];
  __shared__ float vxy[16][20];
  __shared__ float vxx[16][20];

  const int lane = threadIdx.x;
  const int c0 = blockIdx.x * 16;
  const int r0 = blockIdx.y * 16;
  const int p  = blockIdx.z;
  const float* xp = xlr + (size_t)p * HL * WL;
  const float* yp = ylr + (size_t)p * HL * WL;

  // Stage 20x20 halo tile (zero outside image == truncated-window semantics).
  for (int idx = lane; idx < 400; idx += 32) {
    int rr = idx / 20, cc = idx - rr * 20;
    int r = r0 - RAD + rr, c = c0 - RAD + cc;
    float xv = 0.0f, yv = 0.0f;
    if (r >= 0 && r < HL && c >= 0 && c < WL) {
      size_t o = (size_t)r * WL + c;
      xv = xp[o];
      yv = yp[o];
    }
    xs[rr][cc] = xv;
    ys[rr][cc] = yv;
  }
  __syncthreads();

  // Vertical 5-row sums for the 16 output rows over all 20 region columns.
  for (int pos = lane; pos < 320; pos += 32) {
    int i = pos / 20, jj = pos - i * 20;
    float sx = 0.0f, sy = 0.0f, sxy = 0.0f, sxx = 0.0f;
#pragma unroll
    for (int dr = 0; dr < 5; ++dr) {
      float xv = xs[i + dr][jj];
      float yv = ys[i + dr][jj];
      sx += xv; sy += yv; sxy += xv * yv; sxx += xv * xv;
    }
    vx[i][jj] = sx; vy[i][jj] = sy; vxy[i][jj] = sxy; vxx[i][jj] = sxx;
  }
  __syncthreads();

  v8f cx = {0, 0, 0, 0, 0, 0, 0, 0};
  v8f cy = {0, 0, 0, 0, 0, 0, 0, 0};
  v8f cxy = {0, 0, 0, 0, 0, 0, 0, 0};
  v8f cxx = {0, 0, 0, 0, 0, 0, 0, 0};

  const int mm = lane & 15;                 // A-matrix row / B-matrix column
  const int koff = (lane & 16) ? 2 : 0;     // K pair held by upper half-wave

#if HAVE_WMMA_F32
  // Horizontal pass as out(16x16) = V(16x20) x S(20x16), S[k][j] = (j<=k<=j+4).
  // 5 chained K=4 WMMAs per quantity. Exact f32 MACs (ones/zeros in S).
#pragma unroll
  for (int kb = 0; kb < 20; kb += 4) {
    int k0 = kb + koff;
    int k1 = k0 + 1;
    v2f bf;
    bf.x = (k0 >= mm && k0 <= mm + 4) ? 1.0f : 0.0f;
    bf.y = (k1 >= mm && k1 <= mm + 4) ? 1.0f : 0.0f;
    v2f ax; ax.x = vx[mm][k0];  ax.y = vx[mm][k1];
    v2f ay; ay.x = vy[mm][k0];  ay.y = vy[mm][k1];
    v2f az; az.x = vxy[mm][k0]; az.y = vxy[mm][k1];
    v2f aw; aw.x = vxx[mm][k0]; aw.y = vxx[mm][k1];
    cx  = __builtin_amdgcn_wmma_f32_16x16x4_f32(false, ax, false, bf, (short)0, cx,  false, false);
    cy  = __builtin_amdgcn_wmma_f32_16x16x4_f32(false, ay, false, bf, (short)0, cy,  false, false);
    cxy = __builtin_amdgcn_wmma_f32_16x16x4_f32(false, az, false, bf, (short)0, cxy, false, false);
    cxx = __builtin_amdgcn_wmma_f32_16x16x4_f32(false, aw, false, bf, (short)0, cxx, false, false);
  }
#else
  // VALU fallback producing the same C/D register layout.
  {
    const int ibase = (lane & 16) ? 8 : 0;
#pragma unroll
    for (int vr = 0; vr < 8; ++vr) {
      int i = ibase + vr;
      float sx = 0.0f, sy = 0.0f, sxy = 0.0f, sxx = 0.0f;
#pragma unroll
      for (int dc = 0; dc < 5; ++dc) {
        sx += vx[i][mm + dc]; sy += vy[i][mm + dc];
        sxy += vxy[i][mm + dc]; sxx += vxx[i][mm + dc];
      }
      cx[vr] = sx; cy[vr] = sy; cxy[vr] = sxy; cxx[vr] = sxx;
    }
  }
#endif
  (void)koff;

  // Stats -> A, b. Window pixel counts computed analytically (matches
  // box_filter(ones) exactly).
  const int n = c0 + mm;
  const float cntc = (float)(imin_(n + RAD, WL - 1) - imax_(n - RAD, 0) + 1);
  float* Ap = Aout + (size_t)p * HL * WL;
  float* Bp = Bout + (size_t)p * HL * WL;
#pragma unroll
  for (int vr = 0; vr < 8; ++vr) {
    int r = r0 + vr + ((lane & 16) ? 8 : 0);
    float cntr = (float)(imin_(r + RAD, HL - 1) - imax_(r - RAD, 0) + 1);
    float inv = 1.0f / (cntr * cntc);
    float mx = cx[vr] * inv;
    float my = cy[vr] * inv;
    float cov = cxy[vr] * inv - mx * my;
    float var = cxx[vr] * inv - mx * mx;
    float Av = cov / (var + EPSV);
    float Bv = my - Av * mx;
    size_t o = (size_t)r * WL + n;
    Ap[o] = Av;
    Bp[o] = Bv;
  }
}

// ---------------------------------------------------------------------------
// Shared apply: bilinear (align_corners, scale 255/1023 exact integer math)
// upsample of A,b from an 18x18 LDS region + fused multiply-add, float4 NT IO.
// ---------------------------------------------------------------------------
__device__ __forceinline__ void apply_tile(const float* As, const float* Bs,
                                           const float* __restrict__ xp,
                                           float* __restrict__ op,
                                           int r0, int c0, int gy0, int gx0,
                                           int tid) {
  const int tx = tid & 15;
  const int ty = tid >> 4;
  const int cb = c0 + tx * 4;

  int lx0[4], lx1[4];
  float fx[4];
#pragma unroll
  for (int cc = 0; cc < 4; ++cc) {
    int col = cb + cc;
    int t = col * 255;               // src_x = col*255/1023 (align_corners)
    int x0 = t / 1023;
    fx[cc] = (float)(t - x0 * 1023) * (1.0f / 1023.0f);
    int x1 = (x0 + 1 < WL) ? x0 + 1 : WL - 1;
    lx0[cc] = x0 - gx0;
    lx1[cc] = x1 - gx0;
  }

#pragma unroll
  for (int rr = 0; rr < 4; ++rr) {
    int r = r0 + ty + rr * 16;
    int t = r * 255;
    int y0 = t / 1023;
    float fy = (float)(t - y0 * 1023) * (1.0f / 1023.0f);
    int y1 = (y0 + 1 < HL) ? y0 + 1 : HL - 1;
    int row0 = (y0 - gy0) * 18;
    int row1 = (y1 - gy0) * 18;

    const f4* xaddr = (const f4*)(xp + (size_t)r * WH + cb);
    f4 xv = __builtin_nontemporal_load(xaddr);
    f4 ov;
#pragma unroll
    for (int cc = 0; cc < 4; ++cc) {
      float a00 = As[row0 + lx0[cc]], a01 = As[row0 + lx1[cc]];
      float a10 = As[row1 + lx0[cc]], a11 = As[row1 + lx1[cc]];
      float b00 = Bs[row0 + lx0[cc]], b01 = Bs[row0 + lx1[cc]];
      float b10 = Bs[row1 + lx0[cc]], b11 = Bs[row1 + lx1[cc]];
      float at = a00 + (a01 - a00) * fx[cc];
      float ab = a10 + (a11 - a10) * fx[cc];
      float bt = b00 + (b01 - b00) * fx[cc];
      float bb2 = b10 + (b11 - b10) * fx[cc];
      float Av = at + (ab - at) * fy;
      float Bv = bt + (bb2 - bt) * fy;
      ov[cc] = Av * xv[cc] + Bv;
    }
    __builtin_nontemporal_store(ov, (f4*)(op + (size_t)r * WH + cb));
  }
}

// ---------------------------------------------------------------------------
// Kernel 2: 256 threads per 64x64 hi-res tile. Grid (16,16,24).
// ---------------------------------------------------------------------------
__global__ __launch_bounds__(256) void gf_apply(const float* __restrict__ A_ws,
                                                const float* __restrict__ B_ws,
                                                const float* __restrict__ xhr,
                                                float* __restrict__ out) {
  __shared__ float As[18 * 18];
  __shared__ float Bs[18 * 18];

  const int tid = threadIdx.x;
  const int c0 = blockIdx.x * 64;
  const int r0 = blockIdx.y * 64;
  const int p  = blockIdx.z;
  const int gx0 = (c0 * 255) / 1023;
  const int gy0 = (r0 * 255) / 1023;
  const float* Ap = A_ws + (size_t)p * HL * WL;
  const float* Bp = B_ws + (size_t)p * HL * WL;

#if HAVE_ASYNC_LDS
  for (int idx = tid; idx < 18 * 18; idx += 256) {
    int ii = idx / 18, jj = idx - ii * 18;
    int gy = gy0 + ii; if (gy > HL - 1) gy = HL - 1;
    int gx = gx0 + jj; if (gx > WL - 1) gx = WL - 1;
    size_t off = (size_t)gy * WL + gx;
    __builtin_amdgcn_global_load_async_to_lds_b32(
        (__attribute__((address_space(1))) int*)(Ap + off),
        (__attribute__((address_space(3))) int*)&As[idx], 0, 0);
    __builtin_amdgcn_global_load_async_to_lds_b32(
        (__attribute__((address_space(1))) int*)(Bp + off),
        (__attribute__((address_space(3))) int*)&Bs[idx], 0, 0);
  }
  asm volatile("s_wait_asynccnt 0" ::: "memory");
#else
  for (int idx = tid; idx < 18 * 18; idx += 256) {
    int ii = idx / 18, jj = idx - ii * 18;
    int gy = gy0 + ii; if (gy > HL - 1) gy = HL - 1;
    int gx = gx0 + jj; if (gx > WL - 1) gx = WL - 1;
    size_t off = (size_t)gy * WL + gx;
    As[idx] = Ap[off];
    Bs[idx] = Bp[off];
  }
#endif
  __syncthreads();

  apply_tile(As, Bs, xhr + (size_t)p * HH * WH, out + (size_t)p * HH * WH,
             r0, c0, gy0, gx0, tid);
}

// ---------------------------------------------------------------------------
// Fused fallback (used only if d_ws is too small for A/b scratch): recompute
// the 18x18 A/b region per tile directly from the low-res inputs.
// ---------------------------------------------------------------------------
__global__ __launch_bounds__(256) void gf_fused(const float* __restrict__ xlr,
                                                const float* __restrict__ ylr,
                                                const float* __restrict__ xhr,
                                                float* __restrict__ out) {
  __shared__ float xs[22][23];
  __shared__ float ys[22][23];
  __shared__ float vx[18][22];
  __shared__ float vy[18][22];
  __shared__ float vxy[18][22];
  __shared__ float vxx[18][22];
  __shared__ float As[18 * 18];
  __shared__ float Bs[18 * 18];

  const int tid = threadIdx.x;
  const int c0 = blockIdx.x * 64;
  const int r0 = blockIdx.y * 64;
  const int p  = blockIdx.z;
  const int gx0 = (c0 * 255) / 1023;
  const int gy0 = (r0 * 255) / 1023;
  const float* xp = xlr + (size_t)p * HL * WL;
  const float* yp = ylr + (size_t)p * HL * WL;

  for (int idx = tid; idx < 22 * 22; idx += 256) {
    int rr = idx / 22, cc = idx - rr * 22;
    int r = gy0 - RAD + rr, c = gx0 - RAD + cc;
    float xv = 0.0f, yv = 0.0f;
    if (r >= 0 && r < HL && c >= 0 && c < WL) {
      size_t o = (size_t)r * WL + c;
      xv = xp[o];
      yv = yp[o];
    }
    xs[rr][cc] = xv;
    ys[rr][cc] = yv;
  }
  __syncthreads();

  for (int idx = tid; idx < 18 * 22; idx += 256) {
    int ii = idx / 22, cc = idx - ii * 22;
    int rv = imin_(gy0 + ii, HL - 1);
    int base = rv - gy0;  // region row of (rv - RAD)
    float sx = 0.0f, sy = 0.0f, sxy = 0.0f, sxx = 0.0f;
#pragma unroll
    for (int dr = 0; dr < 5; ++dr) {
      float xv = xs[base + dr][cc];
      float yv = ys[base + dr][cc];
      sx += xv; sy += yv; sxy += xv * yv; sxx += xv * xv;
    }
    vx[ii][cc] = sx; vy[ii][cc] = sy; vxy[ii][cc] = sxy; vxx[ii][cc] = sxx;
  }
  __syncthreads();

  for (int idx = tid; idx < 18 * 18; idx += 256) {
    int ii = idx / 18, jj = idx - ii * 18;
    int rv = imin_(gy0 + ii, HL - 1);
    int cv = imin_(gx0 + jj, WL - 1);
    int cbx = cv - gx0;  // region col of (cv - RAD)
    float sx = 0.0f, sy = 0.0f, sxy = 0.0f, sxx = 0.0f;
#pragma unroll
    for (int dc = 0; dc < 5; ++dc) {
      sx += vx[ii][cbx + dc]; sy += vy[ii][cbx + dc];
      sxy += vxy[ii][cbx + dc]; sxx += vxx[ii][cbx + dc];
    }
    float cnt = (float)((imin_(rv + RAD, HL - 1) - imax_(rv - RAD, 0) + 1) *
                        (imin_(cv + RAD, WL - 1) - imax_(cv - RAD, 0) + 1));
    float inv = 1.0f / cnt;
    float mx = sx * inv;
    float my = sy * inv;
    float cov = sxy * inv - mx * my;
    float var = sxx * inv - mx * mx;
    float Av = cov / (var + EPSV);
    As[idx] = Av;
    Bs[idx] = my - Av * mx;
  }
  __syncthreads();

  apply_tile(As, Bs, xhr + (size_t)p * HH * WH, out + (size_t)p * HH * WH,
             r0, c0, gy0, gx0, tid);
}

// ---------------------------------------------------------------------------
extern "C" void kernel_launch(void* const* d_in, const int* in_sizes, int n_in,
                              void* d_out, int out_size, void* d_ws, size_t ws_size,
                              hipStream_t stream) {
  (void)in_sizes; (void)n_in; (void)out_size;
  const float* x_lr = (const float*)d_in[0];
  const float* y_lr = (const float*)d_in[1];
  const float* x_hr = (const float*)d_in[2];
  float* out = (float*)d_out;

  const size_t plane_lr = (size_t)HL * WL;
  const size_t need = 2ull * NPLANE * plane_lr * sizeof(float);

  dim3 gridHi(WH / 64, HH / 64, NPLANE);   // (16,16,24)
  if (ws_size >= need) {
    float* A_ws = (float*)d_ws;
    float* B_ws = A_ws + (size_t)NPLANE * plane_lr;
    dim3 gridLo(WL / 16, HL / 16, NPLANE); // (16,16,24), one wave each
    gf_lowres<<<gridLo, 32, 0, stream>>>(x_lr, y_lr, A_ws, B_ws);
    gf_apply<<<gridHi, 256, 0, stream>>>(A_ws, B_ws, x_hr, out);
  } else {
    gf_fused<<<gridHi, 256, 0, stream>>>(x_lr, y_lr, x_hr, out);
  }
}